// LinearAttention_9096740733676
// MI455X (gfx1250) — compile-verified
//
#include <hip/hip_runtime.h>

typedef __attribute__((ext_vector_type(2))) float v2f;
typedef __attribute__((ext_vector_type(8))) float v8f;

#define N_  8
#define L_  8192
#define S_  8192
#define H_  8
#define D_  64
#define NH_ (N_ * H_)
#define SC  128   // source rows per block, phase 1
#define LC  64    // query rows per block, phase 2

// elu(x)+1 : strictly positive feature map
__device__ __forceinline__ float fm(float x) {
    return x >= 0.0f ? x + 1.0f : __expf(x);
}

// ---------------------------------------------------------------------------
// Phase 1: KV[nh][m][d] = sum_s V[s,m] * fm(K[s,d]);  Ksum[nh][d] = sum_s fm(K[s,d])
// (The reference's /S on V and *S on the output cancel; both are dropped.)
// One block = one (n,h) pair x one 128-row S-chunk. 4 waves, each owns a 16-row
// m-stripe of the 64x64 output and accumulates via v_wmma_f32_16x16x4_f32.
// Partials land in the 1 MB L2-resident workspace via global_atomic_add_f32.
// ---------------------------------------------------------------------------
__global__ __launch_bounds__(128) void la_phase1(const float* __restrict__ keys,
                                                 const float* __restrict__ values,
                                                 float* __restrict__ kv,
                                                 float* __restrict__ ksum) {
    __shared__ float LK[SC][D_];   // fm(K) chunk (fragment reads are row-fixed -> no padding needed)
    __shared__ float LV[SC][D_];   // V chunk

    const int nh = blockIdx.x;
    const int n  = nh / H_;
    const int h  = nh % H_;
    const int s0 = blockIdx.y * SC;
    const int t  = threadIdx.x;

    const size_t rowstride = (size_t)H_ * D_;             // stride between s-rows
    const float* kbase = keys   + ((size_t)n * S_ + s0) * rowstride + (size_t)h * D_;
    const float* vbase = values + ((size_t)n * S_ + s0) * rowstride + (size_t)h * D_;

    // Stage chunk: 2048 float4 over 128 threads, fm applied once here.
    #pragma unroll
    for (int i = 0; i < (SC * D_ / 4) / 128; ++i) {       // 16 iterations
        const int idx4 = t + i * 128;
        const int r = idx4 >> 4;                          // 16 float4 per 64-f32 row
        const int c = (idx4 & 15) * 4;
        const float4 kk = *(const float4*)(kbase + (size_t)r * rowstride + c);
        const float4 vv = *(const float4*)(vbase + (size_t)r * rowstride + c);
        LK[r][c+0] = fm(kk.x); LK[r][c+1] = fm(kk.y);
        LK[r][c+2] = fm(kk.z); LK[r][c+3] = fm(kk.w);
        LV[r][c+0] = vv.x; LV[r][c+1] = vv.y;
        LV[r][c+2] = vv.z; LV[r][c+3] = vv.w;
    }
    __syncthreads();

    // Partial column sums of fm(K) -> Ksum (two threads per column)
    {
        const int d    = t & 63;
        const int half = t >> 6;
        float s = 0.0f;
        #pragma unroll 8
        for (int r = half * 64; r < half * 64 + 64; ++r) s += LK[r][d];
        unsafeAtomicAdd(&ksum[nh * D_ + d], s);
    }

    // C[64(m) x 64(d)] += LV^T(64 x SC) * LK(SC x 64), 16x16x4 f32 WMMA
    const int wave = t >> 5;
    const int lane = t & 31;
    const int sub  = lane & 15;
    const int hi   = lane >> 4;               // lanes 16-31 carry K=2,3
    const int mcol = wave * 16 + sub;         // A-matrix row (m index)

    v8f acc[4] = {};
    for (int s = 0; s < SC; s += 4) {
        const int r0 = s + (hi ? 2 : 0);
        v2f a;
        a.x = LV[r0][mcol];
        a.y = LV[r0 + 1][mcol];
        #pragma unroll
        for (int dj = 0; dj < 4; ++dj) {
            v2f b;
            b.x = LK[r0][dj * 16 + sub];
            b.y = LK[r0 + 1][dj * 16 + sub];
            acc[dj] = __builtin_amdgcn_wmma_f32_16x16x4_f32(
                false, a, false, b, (short)0, acc[dj], false, false);
        }
    }

    // Accumulate tiles into the L2-resident KV workspace.
    float* kvb = kv + (size_t)nh * D_ * D_;
    #pragma unroll
    for (int dj = 0; dj < 4; ++dj) {
        #pragma unroll
        for (int j = 0; j < 8; ++j) {
            const int m = wave * 16 + (hi ? j + 8 : j);   // D-matrix row layout
            const int d = dj * 16 + sub;                  // D-matrix col = lane
            unsafeAtomicAdd(&kvb[m * D_ + d], acc[dj][j]);
        }
    }
}

// ---------------------------------------------------------------------------
// Phase 2: out[l,m] = (1 / (fm(Q[l,:]).Ksum + eps)) * sum_d fm(Q[l,d]) * KV[m,d]
// One block = one (n,h) x 64 query rows. WMMA over k=D in steps of 4.
// LDS rows padded to 65 f32: A/B fragment reads vary by row across lanes.
// ---------------------------------------------------------------------------
__global__ __launch_bounds__(128) void la_phase2(const float* __restrict__ queries,
                                                 const float* __restrict__ kv,
                                                 const float* __restrict__ ksum,
                                                 float* __restrict__ out) {
    __shared__ float LQ[LC][D_ + 1];    // fm(Q), padded (column-wise fragment reads)
    __shared__ float LKV[D_][D_ + 1];   // KV, padded (read transposed as B)
    __shared__ float Ks[D_];
    __shared__ float Zs[LC];

    const int nh = blockIdx.x;
    const int n  = nh / H_;
    const int h  = nh % H_;
    const int l0 = blockIdx.y * LC;
    const int t  = threadIdx.x;

    const size_t rowstride = (size_t)H_ * D_;
    const float* qbase = queries + ((size_t)n * L_ + l0) * rowstride + (size_t)h * D_;

    // Stage fm(Q): 1024 float4 / 128 threads
    #pragma unroll
    for (int i = 0; i < (LC * D_ / 4) / 128; ++i) {       // 8 iterations
        const int idx4 = t + i * 128;
        const int r = idx4 >> 4;
        const int c = (idx4 & 15) * 4;
        const float4 q = *(const float4*)(qbase + (size_t)r * rowstride + c);
        LQ[r][c+0] = fm(q.x); LQ[r][c+1] = fm(q.y);
        LQ[r][c+2] = fm(q.z); LQ[r][c+3] = fm(q.w);
    }
    // Stage KV (hits in L2)
    const float* kvb = kv + (size_t)nh * D_ * D_;
    #pragma unroll
    for (int i = 0; i < (D_ * D_ / 4) / 128; ++i) {       // 8 iterations
        const int idx4 = t + i * 128;
        const int r = idx4 >> 4;
        const int c = (idx4 & 15) * 4;
        const float4 x = *(const float4*)(kvb + r * D_ + c);
        LKV[r][c+0] = x.x; LKV[r][c+1] = x.y; LKV[r][c+2] = x.z; LKV[r][c+3] = x.w;
    }
    if (t < D_) Ks[t] = ksum[nh * D_ + t];
    __syncthreads();

    // Row normalizer: Z = 1 / (fm(q).Ksum + eps)
    if (t < LC) {
        float zdot = 0.0f;
        #pragma unroll
        for (int d = 0; d < D_; ++d) zdot += LQ[t][d] * Ks[d];
        Zs[t] = 1.0f / (zdot + 1e-6f);
    }
    __syncthreads();

    const int wave = t >> 5;
    const int lane = t & 31;
    const int sub  = lane & 15;
    const int hi   = lane >> 4;
    const int lrow = wave * 16 + sub;        // A-matrix row (query row in chunk)

    v8f acc[4] = {};
    #pragma unroll
    for (int k = 0; k < D_; k += 4) {
        const int k0 = k + (hi ? 2 : 0);
        v2f a;
        a.x = LQ[lrow][k0];
        a.y = LQ[lrow][k0 + 1];
        #pragma unroll
        for (int dj = 0; dj < 4; ++dj) {
            v2f b;                            // B[k][m] = KV[m][k]
            b.x = LKV[dj * 16 + sub][k0];
            b.y = LKV[dj * 16 + sub][k0 + 1];
            acc[dj] = __builtin_amdgcn_wmma_f32_16x16x4_f32(
                false, a, false, b, (short)0, acc[dj], false, false);
        }
    }

    // Scale by Z and store: per-VGPR, each half-wave writes 64 contiguous bytes.
    float* obase = out + ((size_t)n * L_ + l0) * rowstride + (size_t)h * D_;
    #pragma unroll
    for (int dj = 0; dj < 4; ++dj) {
        #pragma unroll
        for (int j = 0; j < 8; ++j) {
            const int r = wave * 16 + (hi ? j + 8 : j);
            const int m = dj * 16 + sub;
            obase[(size_t)r * rowstride + m] = acc[dj][j] * Zs[r];
        }
    }
}

// ---------------------------------------------------------------------------
extern "C" void kernel_launch(void* const* d_in, const int* in_sizes, int n_in,
                              void* d_out, int out_size, void* d_ws, size_t ws_size,
                              hipStream_t stream) {
    (void)in_sizes; (void)n_in; (void)out_size; (void)ws_size;
    const float* q = (const float*)d_in[0];
    const float* k = (const float*)d_in[1];
    const float* v = (const float*)d_in[2];
    float* out = (float*)d_out;

    float* kvws   = (float*)d_ws;                         // [NH][64][64]
    float* ksumws = kvws + (size_t)NH_ * D_ * D_;         // [NH][64]
    const size_t zbytes = ((size_t)NH_ * D_ * D_ + (size_t)NH_ * D_) * sizeof(float);

    hipMemsetAsync(d_ws, 0, zbytes, stream);              // capture-legal

    la_phase1<<<dim3(NH_, S_ / SC), 128, 0, stream>>>(k, v, kvws, ksumws);
    la_phase2<<<dim3(NH_, L_ / LC), 128, 0, stream>>>(q, kvws, ksumws, out);
}